// GCM_23390391894819
// MI455X (gfx1250) — compile-verified
//
#include <hip/hip_runtime.h>
#include <math.h>

// ---------------------------------------------------------------------------
// Problem constants (from reference)
// ---------------------------------------------------------------------------
#define BATCH 4
#define CH    16
#define HP    64
#define NP    (HP*HP)     // 4096 tokens per batch at half res
#define NT    (NP/16)     // 256 token tiles per batch
#define H2    128
#define N2    (H2*H2)     // 16384 tokens per batch at full res
#define NT2   (N2/16)     // 1024 tiles per batch
#define EPSV  1e-5f

// ---------------------------------------------------------------------------
// Input index map: recursive dict-insertion order of setup_inputs()
//   x, then params: Q_proj{w,b}, K_proj{w,b}, liner_l{w,b}, linear_o{w,b},
//   linear_p{w,b}, norm{gamma,beta}, sc_h{dw_w,dw_b,pw_w,pw_b,bn{g,b,m,v}},
//   sc_v{...}, conv_h{w,b}, dsc{dw_w,dw_b,pw_w,pw_b,bn{g,b,m,v}}
// ---------------------------------------------------------------------------
enum {
  IN_X = 0,
  IN_QW, IN_QB, IN_KW, IN_KB, IN_LW, IN_LB, IN_OW, IN_OB, IN_PW, IN_PB,
  IN_NG, IN_NB,
  IN_HD_W, IN_HD_B, IN_HP_W, IN_HP_B, IN_HBN_G, IN_HBN_B, IN_HBN_M, IN_HBN_V,
  IN_VD_W, IN_VD_B, IN_VP_W, IN_VP_B, IN_VBN_G, IN_VBN_B, IN_VBN_M, IN_VBN_V,
  IN_CH_W, IN_CH_B,
  IN_DD_W, IN_DD_B, IN_DP_W, IN_DP_B, IN_DBN_G, IN_DBN_B, IN_DBN_M, IN_DBN_V
};

// Workspace layout (float offsets)
#define OFF_XP  0u              // [B,C,64,64]            262144
#define OFF_GAP 262144u         // [B,C]                  64 (padded)
#define OFF_PL  262208u         // [B*N,16] token layout  262144
#define OFF_PH  524352u
#define OFF_XV  786496u
#define OFF_GH  1048640u
#define OFF_GV  1310784u
#define OFF_PN  1572928u        // [B,C,64,64]
#define OFF_UP  1835072u        // [B,C,128,128]          1048576
#define OFF_G3  2883648u        // [B*N2,16]              1048576

typedef float v2f __attribute__((ext_vector_type(2)));
typedef float v8f __attribute__((ext_vector_type(8)));

__device__ __forceinline__ int laneId() { return (int)(threadIdx.x & 31u); }

// D = A(16x16) x B(16x16) + C via 4x V_WMMA_F32_16X16X4_F32.
// Per-lane operand arrays a[8], b[8] use the "split" convention:
//   A: lane L holds row (L&15), channels (8*(L>=16)) .. +8
//   B: lane n holds column (n&15) i.e. token (n&15)'s channels (8*(n>=16)) .. +8
// K-chunk q covers channels {2q, 2q+1, 8+2q, 9+2q} so both operands feed the
// WMMA directly as pairs (v[2q], v[2q+1]) with zero cross-lane movement.
__device__ __forceinline__ v8f wmma16x16(const float* a, const float* b, v8f c) {
#pragma unroll
  for (int q = 0; q < 4; ++q) {
    v2f av; av.x = a[2*q]; av.y = a[2*q+1];
    v2f bv; bv.x = b[2*q]; bv.y = b[2*q+1];
    c = __builtin_amdgcn_wmma_f32_16x16x4_f32(false, av, false, bv, (short)0, c,
                                              false, false);
  }
  return c;
}

__device__ __forceinline__ void load_row8(const float* p, float v[8]) {
  float4 a = ((const float4*)p)[0];
  float4 b = ((const float4*)p)[1];
  v[0]=a.x; v[1]=a.y; v[2]=a.z; v[3]=a.w; v[4]=b.x; v[5]=b.y; v[6]=b.z; v[7]=b.w;
}
__device__ __forceinline__ void store_row8(float* p, const float v[8]) {
  ((float4*)p)[0] = make_float4(v[0],v[1],v[2],v[3]);
  ((float4*)p)[1] = make_float4(v[4],v[5],v[6],v[7]);
}
// tile t of a [*,16] row-major token (or 16x16 weight) buffer
__device__ __forceinline__ void load_tok(const float* buf, int t, float v[8]) {
  int lane = laneId();
  load_row8(buf + ((size_t)(t*16 + (lane&15)))*16 + 8*(lane>>4), v);
}
__device__ __forceinline__ void store_tok(float* buf, int t, const float v[8]) {
  int lane = laneId();
  store_row8(buf + ((size_t)(t*16 + (lane&15)))*16 + 8*(lane>>4), v);
}
// 16-vector (bias / bn param): lane gets elements [8*hi, 8*hi+8)
__device__ __forceinline__ void load_vec16(const float* p, float v[8]) {
  load_row8(p + 8*(laneId()>>4), v);
}
__device__ __forceinline__ float gelu_exact(float v) {
  return 0.5f * v * (1.0f + erff(v * 0.70710678118654752440f));
}
__device__ __forceinline__ float fast_sigmoid(float x) {
  // v_exp_f32 + v_rcp_f32 (avoid the IEEE div_scale/div_fmas sequence)
  return __builtin_amdgcn_rcpf(1.0f + __expf(-x));
}

// ---------------------------------------------------------------------------
// 1) AvgPool2d(2,2): x[B,C,128,128] -> xp[B,C,64,64]
// ---------------------------------------------------------------------------
__global__ void k_pool(const float* __restrict__ x, float* __restrict__ xp) {
  int tid = blockIdx.x * 256 + threadIdx.x;          // 262144
  int xq = tid & 63, yq = (tid >> 6) & 63, bc = tid >> 12;
  const float* s = x + ((size_t)bc)*(128*128) + (yq*2)*128 + xq*2;
  xp[tid] = 0.25f * (s[0] + s[1] + s[128] + s[129]);
}

// ---------------------------------------------------------------------------
// 2) Global average (spatial mean) per (b,c)
// ---------------------------------------------------------------------------
__global__ void k_gap(const float* __restrict__ xp, float* __restrict__ gap) {
  __shared__ float sm[256];
  const float* p = xp + (size_t)blockIdx.x * NP;
  float s = 0.f;
  for (int i = threadIdx.x; i < NP; i += 256) s += p[i];
  sm[threadIdx.x] = s; __syncthreads();
  for (int off = 128; off > 0; off >>= 1) {
    if ((int)threadIdx.x < off) sm[threadIdx.x] += sm[threadIdx.x + off];
    __syncthreads();
  }
  if (threadIdx.x == 0) gap[blockIdx.x] = sm[0] * (1.0f / (float)NP);
}

// ---------------------------------------------------------------------------
// 3) LPG: t = gap*xp tokens; q,k linears; v = sigmoid(q)*k + t; pl = lin(v)
//    xV = lin(xp tokens, linear_o). One wave per 16-token tile. WMMA GEMMs.
// ---------------------------------------------------------------------------
__global__ void k_lpg(const float* __restrict__ xp, const float* __restrict__ gap,
                      const float* __restrict__ Qw, const float* __restrict__ Qb,
                      const float* __restrict__ Kw, const float* __restrict__ Kb,
                      const float* __restrict__ Lw, const float* __restrict__ Lb,
                      const float* __restrict__ Ow, const float* __restrict__ Ob,
                      float* __restrict__ pl, float* __restrict__ xV) {
  int t = blockIdx.x * 8 + (threadIdx.x >> 5);       // 1024 tiles
  int lane = laneId(), col = lane & 15, hi = lane >> 4;
  int b = t / NT, i = t % NT;
  float tokx[8];
#pragma unroll
  for (int r = 0; r < 8; ++r)
    tokx[r] = xp[((size_t)(b*CH + 8*hi + r))*NP + i*16 + col];
  float g[8]; load_vec16(gap + b*CH, g);
  float tg[8];
#pragma unroll
  for (int r = 0; r < 8; ++r) tg[r] = g[r] * tokx[r];

  float wa[8], bias[8]; v8f acc;
  load_tok(Qw, 0, wa); load_vec16(Qb, bias);
#pragma unroll
  for (int r = 0; r < 8; ++r) acc[r] = bias[r];
  acc = wmma16x16(wa, tg, acc);
  float qv[8];
#pragma unroll
  for (int r = 0; r < 8; ++r) qv[r] = acc[r];

  load_tok(Kw, 0, wa); load_vec16(Kb, bias);
#pragma unroll
  for (int r = 0; r < 8; ++r) acc[r] = bias[r];
  acc = wmma16x16(wa, tg, acc);
  float vv[8];
#pragma unroll
  for (int r = 0; r < 8; ++r)
    vv[r] = acc[r] * fast_sigmoid(qv[r]) + tg[r];

  load_tok(Lw, 0, wa); load_vec16(Lb, bias);
#pragma unroll
  for (int r = 0; r < 8; ++r) acc[r] = bias[r];
  acc = wmma16x16(wa, vv, acc);
  float out[8];
#pragma unroll
  for (int r = 0; r < 8; ++r) out[r] = acc[r];
  store_tok(pl, t, out);

  load_tok(Ow, 0, wa); load_vec16(Ob, bias);
#pragma unroll
  for (int r = 0; r < 8; ++r) acc[r] = bias[r];
  acc = wmma16x16(wa, tokx, acc);
#pragma unroll
  for (int r = 0; r < 8; ++r) out[r] = acc[r];
  store_tok(xV, t, out);
}

// ---------------------------------------------------------------------------
// 4) Strip depthwise convs (5x1 and 1x5) + GELU -> token-layout gH, gV
// ---------------------------------------------------------------------------
__global__ void k_strip(const float* __restrict__ xp,
                        const float* __restrict__ wH, const float* __restrict__ bH,
                        const float* __restrict__ wV, const float* __restrict__ bV,
                        float* __restrict__ gH, float* __restrict__ gV) {
  int tid = blockIdx.x * 256 + threadIdx.x;          // 262144
  int c = tid & 15, pix = (tid >> 4) & (NP-1), b = tid >> 16;
  int y = pix >> 6, x = pix & 63;
  const float* plane = xp + ((size_t)(b*CH + c))*NP;
  float aH = bH[c];
#pragma unroll
  for (int k = 0; k < 5; ++k) {
    int yy = y + k - 2;
    if (yy >= 0 && yy < HP) aH += plane[yy*HP + x] * wH[c*5 + k];
  }
  gH[tid] = gelu_exact(aH);
  float aV = bV[c];
#pragma unroll
  for (int k = 0; k < 5; ++k) {
    int xx = x + k - 2;
    if (xx >= 0 && xx < HP) aV += plane[y*HP + xx] * wV[c*5 + k];
  }
  gV[tid] = gelu_exact(aV);
}

// ---------------------------------------------------------------------------
// 5) HPG: xh = BN_h(pw_h(gH)); xv = BN_v(pw_v(gV)); ph = conv_h(xh+xv)
// ---------------------------------------------------------------------------
__global__ void k_hpg(const float* __restrict__ gH, const float* __restrict__ gV,
                      const float* __restrict__ hw, const float* __restrict__ hb,
                      const float* __restrict__ hg, const float* __restrict__ hbe,
                      const float* __restrict__ hm, const float* __restrict__ hv,
                      const float* __restrict__ vw, const float* __restrict__ vb,
                      const float* __restrict__ vg, const float* __restrict__ vbe,
                      const float* __restrict__ vm, const float* __restrict__ vvv,
                      const float* __restrict__ cw, const float* __restrict__ cb,
                      float* __restrict__ ph) {
  int t = blockIdx.x * 8 + (threadIdx.x >> 5);       // 1024 tiles
  float xin[8], wa[8], bias[8]; v8f acc;
  float p0[8], p1[8], p2[8], p3[8], s[8];

  load_tok(gH, t, xin);
  load_tok(hw, 0, wa); load_vec16(hb, bias);
#pragma unroll
  for (int r = 0; r < 8; ++r) acc[r] = bias[r];
  acc = wmma16x16(wa, xin, acc);
  load_vec16(hg, p0); load_vec16(hbe, p1); load_vec16(hm, p2); load_vec16(hv, p3);
#pragma unroll
  for (int r = 0; r < 8; ++r) {
    float sc = p0[r] * rsqrtf(p3[r] + EPSV);
    s[r] = (acc[r] - p2[r]) * sc + p1[r];
  }

  load_tok(gV, t, xin);
  load_tok(vw, 0, wa); load_vec16(vb, bias);
#pragma unroll
  for (int r = 0; r < 8; ++r) acc[r] = bias[r];
  acc = wmma16x16(wa, xin, acc);
  load_vec16(vg, p0); load_vec16(vbe, p1); load_vec16(vm, p2); load_vec16(vvv, p3);
#pragma unroll
  for (int r = 0; r < 8; ++r) {
    float sc = p0[r] * rsqrtf(p3[r] + EPSV);
    s[r] += (acc[r] - p2[r]) * sc + p1[r];
  }

  load_tok(cw, 0, wa); load_vec16(cb, bias);
#pragma unroll
  for (int r = 0; r < 8; ++r) acc[r] = bias[r];
  acc = wmma16x16(wa, s, acc);
  float out[8];
#pragma unroll
  for (int r = 0; r < 8; ++r) out[r] = acc[r];
  store_tok(ph, t, out);
}

// ---------------------------------------------------------------------------
// 6) Fused bidirectional flash attention + linear_p + residual + LayerNorm.
//    One wave owns one 16-row tile; online softmax over 256 column tiles.
//    S^T GEMM puts each softmax row in a lane pair; P tile is directly the
//    next B operand; O^T accumulates in token layout (per-lane corrections).
// ---------------------------------------------------------------------------
__global__ void k_attn(const float* __restrict__ pl, const float* __restrict__ ph,
                       const float* __restrict__ xV,
                       const float* __restrict__ Pw, const float* __restrict__ Pb,
                       const float* __restrict__ ng, const float* __restrict__ nb,
                       float* __restrict__ pn) {
  int t = blockIdx.x * 8 + (threadIdx.x >> 5);       // 1024 tiles
  int lane = laneId(), col = lane & 15, hi = lane >> 4;
  int b = t / NT, i = t % NT;
  const float scale = 0.25f;                         // C^-0.5

  float phi[8], pli[8];
  load_tok(ph, t, phi);
  load_tok(pl, t, pli);

  v8f o1, o2;
#pragma unroll
  for (int r = 0; r < 8; ++r) { o1[r] = 0.f; o2[r] = 0.f; }
  float m1 = -3.0e38f, s1 = 0.f, m2 = -3.0e38f, s2 = 0.f;

  for (int j = 0; j < NT; ++j) {
    int tj = b * NT + j;
    float plj[8], phj[8], xvc[8];
    load_tok(pl, tj, plj);
    load_tok(ph, tj, phj);
#pragma unroll
    for (int k = 0; k < 8; ++k)
      xvc[k] = xV[((size_t)(b*NP + j*16 + k + 8*hi))*16 + col];

    // ---- branch 1: attn_hl rows (ph . pl^T) ----
    v8f st;
#pragma unroll
    for (int r = 0; r < 8; ++r) st[r] = 0.f;
    st = wmma16x16(plj, phi, st);                    // S1^T tile
    float p[8], tm = -3.0e38f;
#pragma unroll
    for (int r = 0; r < 8; ++r) { p[r] = st[r] * scale; tm = fmaxf(tm, p[r]); }
    tm = fmaxf(tm, __shfl_xor(tm, 16));
    float nm = fmaxf(m1, tm);
    float corr = __expf(m1 - nm);
    float ts = 0.f;
#pragma unroll
    for (int r = 0; r < 8; ++r) { p[r] = __expf(p[r] - nm); ts += p[r]; }
    ts += __shfl_xor(ts, 16);
    s1 = s1 * corr + ts; m1 = nm;
#pragma unroll
    for (int r = 0; r < 8; ++r) o1[r] *= corr;
    o1 = wmma16x16(xvc, p, o1);                      // O1^T += xV^T . P^T

    // ---- branch 2: attn_lh rows (pl . ph^T) ----
#pragma unroll
    for (int r = 0; r < 8; ++r) st[r] = 0.f;
    st = wmma16x16(phj, pli, st);
    tm = -3.0e38f;
#pragma unroll
    for (int r = 0; r < 8; ++r) { p[r] = st[r] * scale; tm = fmaxf(tm, p[r]); }
    tm = fmaxf(tm, __shfl_xor(tm, 16));
    nm = fmaxf(m2, tm);
    corr = __expf(m2 - nm);
    ts = 0.f;
#pragma unroll
    for (int r = 0; r < 8; ++r) { p[r] = __expf(p[r] - nm); ts += p[r]; }
    ts += __shfl_xor(ts, 16);
    s2 = s2 * corr + ts; m2 = nm;
#pragma unroll
    for (int r = 0; r < 8; ++r) o2[r] *= corr;
    o2 = wmma16x16(xvc, p, o2);
  }

  float inv1 = __builtin_amdgcn_rcpf(s1), inv2 = __builtin_amdgcn_rcpf(s2);
  float pr[8];
#pragma unroll
  for (int r = 0; r < 8; ++r) pr[r] = o1[r]*inv1 + o2[r]*inv2;

  // linear_p + residual xV
  float wa[8], bias[8]; v8f acc;
  load_tok(Pw, 0, wa); load_vec16(Pb, bias);
#pragma unroll
  for (int r = 0; r < 8; ++r) acc[r] = bias[r];
  acc = wmma16x16(wa, pr, acc);
  float xvi[8]; load_tok(xV, t, xvi);
  float pv[8];
#pragma unroll
  for (int r = 0; r < 8; ++r) pv[r] = acc[r] + xvi[r];

  // LayerNorm over 16 channels per token (lane pair holds one token)
  float sum = 0.f;
#pragma unroll
  for (int r = 0; r < 8; ++r) sum += pv[r];
  sum += __shfl_xor(sum, 16);
  float mu = sum * (1.f/16.f);
  float vs = 0.f;
#pragma unroll
  for (int r = 0; r < 8; ++r) { float d = pv[r]-mu; vs += d*d; }
  vs += __shfl_xor(vs, 16);
  float rstd = rsqrtf(vs * (1.f/16.f) + EPSV);
  float gg[8], bb[8]; load_vec16(ng, gg); load_vec16(nb, bb);
#pragma unroll
  for (int r = 0; r < 8; ++r) {
    float val = (pv[r]-mu)*rstd*gg[r] + bb[r];
    pn[((size_t)(b*CH + 8*hi + r))*NP + i*16 + col] = val;   // [B,C,64,64]
  }
}

// ---------------------------------------------------------------------------
// 7) Bilinear x2 upsample, align_corners=True: pn[B,C,64,64] -> up[B,C,128,128]
// ---------------------------------------------------------------------------
__global__ void k_up(const float* __restrict__ pn, float* __restrict__ up) {
  int tid = blockIdx.x * 256 + threadIdx.x;          // 1048576
  int x = tid & 127, y = (tid >> 7) & 127, bc = tid >> 14;
  float fy = y * (63.0f/127.0f); int y0 = (int)fy; float wy = fy - (float)y0;
  int y1 = min(y0 + 1, 63);
  float fx = x * (63.0f/127.0f); int x0 = (int)fx; float wx = fx - (float)x0;
  int x1 = min(x0 + 1, 63);
  const float* p = pn + (size_t)bc * NP;
  float v00 = p[y0*64+x0], v01 = p[y0*64+x1], v10 = p[y1*64+x0], v11 = p[y1*64+x1];
  up[tid] = (1.f-wy)*((1.f-wx)*v00 + wx*v01) + wy*((1.f-wx)*v10 + wx*v11);
}

// ---------------------------------------------------------------------------
// 8) dsc depthwise 3x3 conv + GELU -> token-layout g3
// ---------------------------------------------------------------------------
__global__ void k_dw3(const float* __restrict__ up, const float* __restrict__ w,
                      const float* __restrict__ bvec, float* __restrict__ g3) {
  int tid = blockIdx.x * 256 + threadIdx.x;          // 1048576
  int c = tid & 15, pix = (tid >> 4) & (N2-1), b = tid >> 18;
  int y = pix >> 7, x = pix & 127;
  const float* plane = up + ((size_t)(b*CH + c))*N2;
  float a = bvec[c];
#pragma unroll
  for (int ky = 0; ky < 3; ++ky) {
    int yy = y + ky - 1;
    if (yy < 0 || yy >= H2) continue;
#pragma unroll
    for (int kx = 0; kx < 3; ++kx) {
      int xx = x + kx - 1;
      if (xx < 0 || xx >= H2) continue;
      a += plane[yy*H2 + xx] * w[c*9 + ky*3 + kx];
    }
  }
  g3[tid] = gelu_exact(a);
}

// ---------------------------------------------------------------------------
// 9) dsc pointwise conv (WMMA) + BN + residual -> out[B,C,128,128]
// ---------------------------------------------------------------------------
__global__ void k_final(const float* __restrict__ g3, const float* __restrict__ up,
                        const float* __restrict__ pw, const float* __restrict__ pb,
                        const float* __restrict__ bg, const float* __restrict__ bbt,
                        const float* __restrict__ bm, const float* __restrict__ bv,
                        float* __restrict__ out) {
  int t = blockIdx.x * 8 + (threadIdx.x >> 5);       // 4096 tiles
  int lane = laneId(), col = lane & 15, hi = lane >> 4;
  int b = t >> 10, i = t & 1023;
  float xin[8], wa[8], bias[8]; v8f acc;
  load_tok(g3, t, xin);
  load_tok(pw, 0, wa); load_vec16(pb, bias);
#pragma unroll
  for (int r = 0; r < 8; ++r) acc[r] = bias[r];
  acc = wmma16x16(wa, xin, acc);
  float g[8], be[8], mm[8], vv[8];
  load_vec16(bg, g); load_vec16(bbt, be); load_vec16(bm, mm); load_vec16(bv, vv);
#pragma unroll
  for (int r = 0; r < 8; ++r) {
    size_t idx = ((size_t)(b*CH + 8*hi + r))*N2 + i*16 + col;
    float sc = g[r] * rsqrtf(vv[r] + EPSV);
    out[idx] = (acc[r] - mm[r]) * sc + be[r] + up[idx];
  }
}

// ---------------------------------------------------------------------------
extern "C" void kernel_launch(void* const* d_in, const int* in_sizes, int n_in,
                              void* d_out, int out_size, void* d_ws, size_t ws_size,
                              hipStream_t stream) {
  (void)in_sizes; (void)n_in; (void)out_size; (void)ws_size;
  const float* P[40];
  for (int k = 0; k < 39; ++k) P[k] = (const float*)d_in[k];
  float* ws  = (float*)d_ws;
  float* xp  = ws + OFF_XP;
  float* gap = ws + OFF_GAP;
  float* pl  = ws + OFF_PL;
  float* ph  = ws + OFF_PH;
  float* xV  = ws + OFF_XV;
  float* gH  = ws + OFF_GH;
  float* gV  = ws + OFF_GV;
  float* pn  = ws + OFF_PN;
  float* up  = ws + OFF_UP;
  float* g3  = ws + OFF_G3;
  float* out = (float*)d_out;

  k_pool <<<1024, 256, 0, stream>>>(P[IN_X], xp);
  k_gap  <<<64,   256, 0, stream>>>(xp, gap);
  k_lpg  <<<128,  256, 0, stream>>>(xp, gap,
            P[IN_QW], P[IN_QB], P[IN_KW], P[IN_KB],
            P[IN_LW], P[IN_LB], P[IN_OW], P[IN_OB], pl, xV);
  k_strip<<<1024, 256, 0, stream>>>(xp, P[IN_HD_W], P[IN_HD_B],
            P[IN_VD_W], P[IN_VD_B], gH, gV);
  k_hpg  <<<128,  256, 0, stream>>>(gH, gV,
            P[IN_HP_W], P[IN_HP_B], P[IN_HBN_G], P[IN_HBN_B], P[IN_HBN_M], P[IN_HBN_V],
            P[IN_VP_W], P[IN_VP_B], P[IN_VBN_G], P[IN_VBN_B], P[IN_VBN_M], P[IN_VBN_V],
            P[IN_CH_W], P[IN_CH_B], ph);
  k_attn <<<128,  256, 0, stream>>>(pl, ph, xV,
            P[IN_PW], P[IN_PB], P[IN_NG], P[IN_NB], pn);
  k_up   <<<4096, 256, 0, stream>>>(pn, up);
  k_dw3  <<<4096, 256, 0, stream>>>(up, P[IN_DD_W], P[IN_DD_B], g3);
  k_final<<<512,  256, 0, stream>>>(g3, up,
            P[IN_DP_W], P[IN_DP_B], P[IN_DBN_G], P[IN_DBN_B], P[IN_DBN_M], P[IN_DBN_V],
            out);
}